// LearnableSparseLinear_9388798509060
// MI455X (gfx1250) — compile-verified
//
#include <hip/hip_runtime.h>
#include <stdint.h>

typedef __attribute__((ext_vector_type(2))) float v2f;
typedef __attribute__((ext_vector_type(8))) float v8f;

#define B_TILE       16
#define IN_DIM       4096
#define OUT_DIM      2048
#define KK           32
#define LDS_STRIDE   4100   /* dwords per staged row; 4100 % 64 == 4 -> 16 rows hit 16 distinct banks */
#define THREADS      512    /* 16 waves of 32 */
#define CH_SPLIT     2
#define OUT_PER_BLK  (OUT_DIM / CH_SPLIT)          /* 1024 */
#define CH_PER_WAVE  (OUT_PER_BLK / (THREADS/32))  /* 64   */
#define OGROUPS      (CH_PER_WAVE / 16)            /* 4    */

__global__ __launch_bounds__(THREADS)
void sparse_linear_wmma(const float* __restrict__ x,
                        const float* __restrict__ values,
                        const int*   __restrict__ cols,
                        float* __restrict__ out) {
  extern __shared__ float lds[];

  const int tid   = threadIdx.x;
  const int lane  = tid & 31;
  /* wave index is wave-uniform: force it into an SGPR so cols/vals/out
     addressing becomes scalar-base + 32-bit voffset (saddr form).       */
  const int wavei = __builtin_amdgcn_readfirstlane(tid >> 5);
  const int b0    = blockIdx.x * B_TILE;
  const int khalf = lane >> 4;   /* 0: lanes 0-15 (K pair 0,1), 1: lanes 16-31 (K pair 2,3) */
  const int nlan  = lane & 15;   /* A row (M) for this lane; also C column (N)              */

  /* ---- stage 16 rows of x into LDS via CDNA5 async global->LDS copies ---- */
  /* 16*4096 floats = 16384 float4 = 512 threads * 32 iterations              */
  for (int i = 0; i < 32; ++i) {
    int idx = tid + i * THREADS;   /* float4 index within the tile */
    int r   = idx >> 10;           /* 1024 float4 per row          */
    int c4  = idx & 1023;
    unsigned goff = (unsigned)(((b0 + r) * IN_DIM + c4 * 4) * 4); /* byte off from x   */
    unsigned loff = (unsigned)((r * LDS_STRIDE + c4 * 4) * 4);    /* byte off into LDS */
    asm volatile("global_load_async_to_lds_b128 %0, %1, %2"
                 :
                 : "v"(loff), "v"(goff), "s"(x)
                 : "memory");
  }
  asm volatile("s_wait_asynccnt 0" ::: "memory");
  __syncthreads();

  const int ldsrow = nlan * LDS_STRIDE;   /* lane's gather base (dwords) */

  /* ---- per-wave: 4 groups of 16 output channels, 16x16 C tile each ---- */
  for (int og = 0; og < OGROUPS; ++og) {
    const int obase = blockIdx.y * OUT_PER_BLK + wavei * CH_PER_WAVE + og * 16; /* scalar */

    /* Hoist this lane's B-column values: channel (obase+nlan), this lane's K slots.
       B f32 4x16 layout: vgpr0 holds K = 2*khalf, vgpr1 holds K = 2*khalf+1.
       Scalar base (values + obase*KK) + per-lane offset -> saddr-form loads.      */
    v2f bf[8];
    const float* vbase = values + (size_t)obase * KK;   /* scalar */
    const int    voff  = nlan * KK + 2 * khalf;         /* per-lane */
#pragma unroll
    for (int s = 0; s < 8; ++s)
      bf[s] = *(const v2f*)(vbase + voff + 4 * s);

    /* Two independent accumulation chains (even/odd channel) so the XDL pipe
       always has an independent WMMA available despite the serial C += chain. */
    v8f c0 = {};
    v8f c1 = {};

    for (int n = 0; n < 16; n += 2) {        /* one channel -> one C column */
      const int* cbase0 = cols + (size_t)(obase + n) * KK;       /* scalar */
      const int* cbase1 = cols + (size_t)(obase + n + 1) * KK;   /* scalar */
      const int  coff   = 2 * khalf;                             /* per-lane */
      const bool mine0  = (nlan == n);
      const bool mine1  = (nlan == n + 1);
#pragma unroll
      for (int s = 0; s < 8; ++s) {          /* K slices of 4 */
        int2 cc0 = *(const int2*)(cbase0 + coff + 4 * s);
        int2 cc1 = *(const int2*)(cbase1 + coff + 4 * s);
        v2f a0, a1;  /* A f32 16x4: lane row = nlan, K pair (2*khalf, 2*khalf+1) */
        a0.x = lds[ldsrow + cc0.x];
        a0.y = lds[ldsrow + cc0.y];
        a1.x = lds[ldsrow + cc1.x];
        a1.y = lds[ldsrow + cc1.y];
        v2f b0, b1;  /* block-diagonal B: only column n carries this channel's vals */
        b0.x = mine0 ? bf[s].x : 0.0f;
        b0.y = mine0 ? bf[s].y : 0.0f;
        b1.x = mine1 ? bf[s].x : 0.0f;
        b1.y = mine1 ? bf[s].y : 0.0f;
        c0 = __builtin_amdgcn_wmma_f32_16x16x4_f32(false, a0, false, b0,
                                                   (short)0, c0, false, false);
        c1 = __builtin_amdgcn_wmma_f32_16x16x4_f32(false, a1, false, b1,
                                                   (short)0, c1, false, false);
      }
    }

    v8f c = c0 + c1;

    /* C/D 16x16 f32 layout: vgpr r -> M = r + 8*khalf, N = nlan.
       out is write-once: non-temporal stores keep x/cols/vals hot in L2. */
    float* obase_p = out + (size_t)b0 * OUT_DIM + obase;  /* scalar */
#pragma unroll
    for (int r = 0; r < 8; ++r) {
      int mm = khalf * 8 + r;
      __builtin_nontemporal_store(c[r], obase_p + (size_t)mm * OUT_DIM + nlan);
    }
  }
}

extern "C" void kernel_launch(void* const* d_in, const int* in_sizes, int n_in,
                              void* d_out, int out_size, void* d_ws, size_t ws_size,
                              hipStream_t stream) {
  (void)in_sizes; (void)n_in; (void)out_size; (void)d_ws; (void)ws_size;
  const float* x      = (const float*)d_in[0];
  const float* values = (const float*)d_in[1];
  const int*   cols   = (const int*)d_in[2];
  float*       out    = (float*)d_out;

  const size_t shmem = (size_t)B_TILE * LDS_STRIDE * sizeof(float); /* 262400 B */
  hipFuncSetAttribute((const void*)sparse_linear_wmma,
                      hipFuncAttributeMaxDynamicSharedMemorySize, (int)shmem);

  dim3 grid(4096 / B_TILE, CH_SPLIT);   /* (256, 2) */
  sparse_linear_wmma<<<grid, THREADS, shmem, stream>>>(x, values, cols, out);
}